// SGRNModel_944892805447
// MI455X (gfx1250) — compile-verified
//
#include <hip/hip_runtime.h>
#include <stdint.h>

#define N_NODES 8192
#define SEQ     128
#define EMB     256
#define LAT     128
#define HVGS    2000
#define NEDGE   131072

typedef float v2f __attribute__((ext_vector_type(2)));
typedef float v8f __attribute__((ext_vector_type(8)));

#define V8ZERO {0.0f,0.0f,0.0f,0.0f,0.0f,0.0f,0.0f,0.0f}

// D = A(16x4) * B(4x16) + C, fp32 WMMA (native CDNA5 fp32 matrix op)
__device__ __forceinline__ v8f wmma_f32(v2f a, v2f b, v8f c) {
  return __builtin_amdgcn_wmma_f32_16x16x4_f32(false, a, false, b, (short)0, c,
                                               false, false);
}

// CDNA5 async memory->LDS copy (ASYNCcnt-tracked); inline asm per
// CDNA5 ISA so it is portable across toolchains.
// VDST operand = LDS byte offset (dynamic-LDS base is 0: no static LDS here).
__device__ __forceinline__ void async_ld_b128(uint32_t lds_byte_off,
                                              const void* gaddr) {
  asm volatile("global_load_async_to_lds_b128 %0, %1, off"
               :: "v"(lds_byte_off), "v"(gaddr) : "memory");
}
__device__ __forceinline__ void wait_async0() {
  asm volatile("s_wait_asynccnt 0x0" ::: "memory");
}

// A fragment (16x4 f32): element (m,k) -> lane = m + 16*(k>>1), reg = k&1
__device__ __forceinline__ v2f load_a_f32(const float* base, int pitch, int m0,
                                          int k0, int lane) {
  const int m  = m0 + (lane & 15);
  const int kk = k0 + ((lane >> 4) << 1);
  v2f a;
  a.x = base[(size_t)m * pitch + kk];
  a.y = base[(size_t)m * pitch + kk + 1];
  return a;
}

// B fragment (4x16 f32) from row-major W[k][n], leading dim ldn
__device__ __forceinline__ v2f load_b_glb(const float* base, int ldn, int k0,
                                          int n0, int lane) {
  const int n  = n0 + (lane & 15);
  const int kk = k0 + ((lane >> 4) << 1);
  v2f b;
  b.x = base[(size_t)kk * ldn + n];
  b.y = base[(size_t)(kk + 1) * ldn + n];
  return b;
}

// B fragment from LDS, row-major M[k][n] (pitch in floats)
__device__ __forceinline__ v2f load_b_lds(const float* base, int pitch, int k0,
                                          int n0, int lane) {
  const int n  = n0 + (lane & 15);
  const int kk = k0 + ((lane >> 4) << 1);
  v2f b;
  b.x = base[kk * pitch + n];
  b.y = base[(kk + 1) * pitch + n];
  return b;
}

// B fragment for Kt: B[k][n] = M[n][k] (transposed read; pitch 129 -> conflict-free)
__device__ __forceinline__ v2f load_bT_lds(const float* base, int pitch, int k0,
                                           int n0, int lane) {
  const int n  = n0 + (lane & 15);
  const int kk = k0 + ((lane >> 4) << 1);
  v2f b;
  b.x = base[n * pitch + kk];
  b.y = base[n * pitch + kk + 1];
  return b;
}

// ---------------- Phase 1: fused per-node attention -----------------------
// LDS layout (floats): emb [128][260] @0 (aliased by V [128][129]@0 and
// att [128][129]@16512 once emb is dead), Q @33280, K @49792, zbuf @66304.
#define EP 260
#define QP 129
#define OFF_ATT  (128 * QP)
#define OFF_Q    (128 * EP)
#define OFF_K    (OFF_Q + 128 * QP)
#define OFF_Z    (OFF_K + 128 * QP)
#define LDS1_FLOATS (OFF_Z + 128)

__global__ __launch_bounds__(256)
void attn_fused(const float* __restrict__ emb,
                const float* __restrict__ Wq, const float* __restrict__ bq,
                const float* __restrict__ Wk, const float* __restrict__ bk,
                const float* __restrict__ Wv, const float* __restrict__ bv,
                const float* __restrict__ Wo, const float* __restrict__ bo,
                float* __restrict__ attg, float* __restrict__ zg) {
  extern __shared__ float smem[];
  float* embF = smem;            // [128][EP]
  float* Vb   = smem;            // [128][QP] (aliases emb after barrier)
  float* Ab   = smem + OFF_ATT;  // [128][QP] (aliases emb tail)
  float* Qb   = smem + OFF_Q;    // [128][QP]
  float* Kb   = smem + OFF_K;    // [128][QP]
  float* zb   = smem + OFF_Z;    // [128]

  const int node = blockIdx.x;
  const int tid  = threadIdx.x;
  const int wave = tid >> 5;        // 8 waves of 32 (wave32)
  const int lane = tid & 31;
  const int m0   = wave * 16;       // this wave owns rows m0..m0+15
  const int nl   = lane & 15;
  const int mb   = m0 + ((lane >> 4) << 3);

  if (tid < LAT) zb[tid] = 0.0f;

  // Stage emb[node] (128x256 f32 = 128 KB) into LDS with CDNA5 async copies:
  // 8192 x b128, no VGPR round-trip, tracked by ASYNCcnt.
  {
    const float* esrc = emb + (size_t)node * SEQ * EMB;
    for (int i = tid; i < SEQ * EMB / 4; i += 256) {
      const int r  = i >> 6;          // 64 float4 per row
      const int c4 = (i & 63) << 2;
      async_ld_b128((uint32_t)((r * EP + c4) * sizeof(float)),
                    (const void*)(esrc + ((size_t)i << 2)));
    }
    // While async copies are in flight, warm L2 with the (L2-resident) weights.
    for (int i = tid * 32; i < EMB * LAT; i += 256 * 32) {
      __builtin_prefetch(Wq + i, 0, 3);   // global_prefetch_b8
      __builtin_prefetch(Wk + i, 0, 3);
      __builtin_prefetch(Wv + i, 0, 3);
    }
    for (int i = tid * 32; i < LAT * LAT; i += 256 * 32)
      __builtin_prefetch(Wo + i, 0, 3);
    wait_async0();                        // s_wait_asynccnt 0 (own wave's copies)
  }
  __syncthreads();                        // all waves' copies visible

  // ---- Q = emb@Wq + bq and K = emb@Wk + bk (shared A fragments) ----
  {
    v8f q[8], k[8];
#pragma unroll
    for (int t = 0; t < 8; ++t) { q[t] = (v8f)V8ZERO; k[t] = (v8f)V8ZERO; }
    for (int k0 = 0; k0 < EMB; k0 += 4) {
      const v2f a = load_a_f32(embF, EP, m0, k0, lane);
#pragma unroll
      for (int t = 0; t < 8; ++t) {
        q[t] = wmma_f32(a, load_b_glb(Wq, LAT, k0, t * 16, lane), q[t]);
        k[t] = wmma_f32(a, load_b_glb(Wk, LAT, k0, t * 16, lane), k[t]);
      }
    }
#pragma unroll
    for (int t = 0; t < 8; ++t) {
      const int n = t * 16 + nl;
      const float bqv = bq[n], bkv = bk[n];
#pragma unroll
      for (int r = 0; r < 8; ++r) {
        Qb[(mb + r) * QP + n] = q[t][r] + bqv;
        Kb[(mb + r) * QP + n] = k[t][r] + bkv;
      }
    }
  }

  // ---- V = emb@Wv + bv: compute into regs, then overwrite emb region ----
  {
    v8f vv[8];
#pragma unroll
    for (int t = 0; t < 8; ++t) vv[t] = (v8f)V8ZERO;
    for (int k0 = 0; k0 < EMB; k0 += 4) {
      const v2f a = load_a_f32(embF, EP, m0, k0, lane);
#pragma unroll
      for (int t = 0; t < 8; ++t)
        vv[t] = wmma_f32(a, load_b_glb(Wv, LAT, k0, t * 16, lane), vv[t]);
    }
    __syncthreads();  // all emb reads (Q,K,V) complete before aliasing write
#pragma unroll
    for (int t = 0; t < 8; ++t) {
      const int n = t * 16 + nl;
      const float bvv = bv[n];
#pragma unroll
      for (int r = 0; r < 8; ++r) Vb[(mb + r) * QP + n] = vv[t][r] + bvv;
    }
    __syncthreads();  // V and K visible to all waves
  }

  // ---- att = tanh(Q @ K^T * 1/sqrt(EMB)) -> global + LDS ----
  {
    v8f aw[8];
#pragma unroll
    for (int t = 0; t < 8; ++t) aw[t] = (v8f)V8ZERO;
    for (int k0 = 0; k0 < LAT; k0 += 4) {
      const v2f a = load_a_f32(Qb, QP, m0, k0, lane);
#pragma unroll
      for (int t = 0; t < 8; ++t)
        aw[t] = wmma_f32(a, load_bT_lds(Kb, QP, k0, t * 16, lane), aw[t]);
    }
    float* ag = attg + (size_t)node * SEQ * SEQ;
#pragma unroll
    for (int t = 0; t < 8; ++t) {
      const int n = t * 16 + nl;
#pragma unroll
      for (int r = 0; r < 8; ++r) {
        const float w = tanhf(aw[t][r] * 0.0625f);
        Ab[(mb + r) * QP + n] = w;                       // own rows only
        ag[(size_t)(mb + r) * SEQ + n] = w;
      }
    }
  }
  // No barrier: each wave consumes only its own att rows below.

  // ---- att_out = att @ V (own rows; result overwrites own Q rows) ----
  {
    v8f ao[8];
#pragma unroll
    for (int t = 0; t < 8; ++t) ao[t] = (v8f)V8ZERO;
    for (int k0 = 0; k0 < SEQ; k0 += 4) {
      const v2f a = load_a_f32(Ab, QP, m0, k0, lane);
#pragma unroll
      for (int t = 0; t < 8; ++t)
        ao[t] = wmma_f32(a, load_b_lds(Vb, QP, k0, t * 16, lane), ao[t]);
    }
#pragma unroll
    for (int t = 0; t < 8; ++t) {
      const int n = t * 16 + nl;
#pragma unroll
      for (int r = 0; r < 8; ++r) Qb[(mb + r) * QP + n] = ao[t][r];
    }
  }

  // ---- y = att_out @ Wo ; z = mean_s(y) + bo ----
  {
    v8f y[8];
#pragma unroll
    for (int t = 0; t < 8; ++t) y[t] = (v8f)V8ZERO;
    for (int k0 = 0; k0 < LAT; k0 += 4) {
      const v2f a = load_a_f32(Qb, QP, m0, k0, lane);
#pragma unroll
      for (int t = 0; t < 8; ++t)
        y[t] = wmma_f32(a, load_b_glb(Wo, LAT, k0, t * 16, lane), y[t]);
    }
#pragma unroll
    for (int t = 0; t < 8; ++t) {
      float p = 0.0f;
#pragma unroll
      for (int r = 0; r < 8; ++r) p += y[t][r];
      p += __shfl_xor(p, 16, 32);           // fold the two row-halves (wave32)
      if (lane < 16) atomicAdd(&zb[t * 16 + nl], p);   // ds_add_f32
    }
  }
  __syncthreads();
  if (tid < LAT)
    zg[(size_t)node * LAT + tid] = zb[tid] * (1.0f / 128.0f) + bo[tid];
}

// ---------------- Phase 2: GCN ---------------------------------------------
__global__ void k_deg_init(float* deg) {
  const int i = blockIdx.x * 256 + threadIdx.x;
  if (i < N_NODES) deg[i] = 1.0f;  // self loop
}

__global__ void k_deg_count(const long long* __restrict__ ei, float* deg) {
  const int e = blockIdx.x * 256 + threadIdx.x;
  if (e < NEDGE) atomicAdd(&deg[(int)ei[NEDGE + e]], 1.0f);
}

__global__ void k_dinv(const float* __restrict__ deg, float* dinv) {
  const int i = blockIdx.x * 256 + threadIdx.x;
  if (i < N_NODES) dinv[i] = rsqrtf(deg[i]);
}

// h = z @ Wg  (8192x128 @ 128x2000), WMMA with z tile staged in LDS
#define ZPITCH 132
__global__ __launch_bounds__(256)
void k_hgemm(const float* __restrict__ z, const float* __restrict__ Wg,
             float* __restrict__ h) {
  __shared__ float zt[16 * ZPITCH];
  const int row0 = blockIdx.x * 16;
  const int tid  = threadIdx.x;
  const int wave = tid >> 5;
  const int lane = tid & 31;
  for (int i = tid; i < 16 * LAT; i += 256)
    zt[(i >> 7) * ZPITCH + (i & 127)] = z[(size_t)(row0 + (i >> 7)) * LAT + (i & 127)];
  __syncthreads();
  for (int t = wave; t < HVGS / 16; t += 8) {
    v8f acc = (v8f)V8ZERO;
    for (int k0 = 0; k0 < LAT; k0 += 4)
      acc = wmma_f32(load_a_f32(zt, ZPITCH, 0, k0, lane),
                     load_b_glb(Wg, HVGS, k0, t * 16, lane), acc);
    const int n  = t * 16 + (lane & 15);
    const int mr = row0 + ((lane >> 4) << 3);
#pragma unroll
    for (int r = 0; r < 8; ++r) h[(size_t)(mr + r) * HVGS + n] = acc[r];
  }
}

// out = dinv[n]^2 * h[n] + bg  (self-loop term; also initializes out)
__global__ void k_selfbias(const float* __restrict__ h,
                           const float* __restrict__ dinv,
                           const float* __restrict__ bg,
                           float* __restrict__ out) {
  const size_t i = (size_t)blockIdx.x * 256 + threadIdx.x;
  if (i < (size_t)N_NODES * HVGS) {
    const int n = (int)(i / HVGS);
    const int c = (int)(i % HVGS);
    out[i] = dinv[n] * dinv[n] * h[i] + bg[c];
  }
}

// out[dst] += dinv[src]*dinv[dst]*h[src] for each edge
__global__ void k_scatter(const long long* __restrict__ ei,
                          const float* __restrict__ h,
                          const float* __restrict__ dinv,
                          float* __restrict__ out) {
  const int e = blockIdx.x;
  const int s = (int)ei[e];
  const int d = (int)ei[NEDGE + e];
  const float norm = dinv[s] * dinv[d];
  const float* hs = h + (size_t)s * HVGS;
  float* od = out + (size_t)d * HVGS;
  for (int c = threadIdx.x; c < HVGS; c += 256)
    atomicAdd(&od[c], norm * hs[c]);  // global_atomic_add_f32
}

// ---------------------------------------------------------------------------
extern "C" void kernel_launch(void* const* d_in, const int* in_sizes, int n_in,
                              void* d_out, int out_size, void* d_ws,
                              size_t ws_size, hipStream_t stream) {
  (void)in_sizes; (void)n_in; (void)out_size; (void)ws_size;
  const float*     emb = (const float*)d_in[0];
  const long long* ei  = (const long long*)d_in[1];  // int64 edge_index [2,E]
  const float* Wq = (const float*)d_in[2];
  const float* bq = (const float*)d_in[3];
  const float* Wk = (const float*)d_in[4];
  const float* bk = (const float*)d_in[5];
  const float* Wv = (const float*)d_in[6];
  const float* bv = (const float*)d_in[7];
  const float* Wo = (const float*)d_in[8];
  const float* bo = (const float*)d_in[9];
  const float* Wg = (const float*)d_in[10];
  const float* bg = (const float*)d_in[11];

  float* out  = (float*)d_out;
  float* attg = out;                                   // [N,S,S]
  float* zg   = out + (size_t)N_NODES * SEQ * SEQ;     // [N,L]
  float* gout = zg + (size_t)N_NODES * LAT;            // [N,HVGS]

  float* deg  = (float*)d_ws;          // N floats
  float* dinv = deg + N_NODES;         // N floats
  float* h    = dinv + N_NODES;        // N*HVGS floats (64 MB scratch)

  const int lds1 = LDS1_FLOATS * (int)sizeof(float);   // 265,728 B (< 320 KB)
  (void)hipFuncSetAttribute((const void*)attn_fused,
                            hipFuncAttributeMaxDynamicSharedMemorySize, lds1);

  attn_fused<<<N_NODES, 256, lds1, stream>>>(emb, Wq, bq, Wk, bk, Wv, bv, Wo,
                                             bo, attg, zg);

  k_deg_init<<<(N_NODES + 255) / 256, 256, 0, stream>>>(deg);
  k_deg_count<<<(NEDGE + 255) / 256, 256, 0, stream>>>(ei, deg);
  k_dinv<<<(N_NODES + 255) / 256, 256, 0, stream>>>(deg, dinv);

  k_hgemm<<<N_NODES / 16, 256, 0, stream>>>(zg, Wg, h);

  const long long tot = (long long)N_NODES * HVGS;
  k_selfbias<<<(int)((tot + 255) / 256), 256, 0, stream>>>(h, dinv, bg, gout);
  k_scatter<<<NEDGE, 256, 0, stream>>>(ei, h, dinv, gout);
}